// GraphConvolutionalNetwork_2697239461977
// MI455X (gfx1250) — compile-verified
//
#include <hip/hip_runtime.h>

#define N_NODES   100000
#define N_EDGES   1250000
#define D         64
#define N_CLASSES 40
#define TILES     (N_NODES / 16)   // 6250 exactly, no tail

typedef float v2f __attribute__((ext_vector_type(2)));
typedef float v8f __attribute__((ext_vector_type(8)));

// ---------------------------------------------------------------------------
// One 16x16 output tile of  C = A(16xK) * W(K x ldw, cols [n0,n0+16)) + cinit
// using V_WMMA_F32_16X16X4_F32, accumulating over K in steps of 4.
// Fragment layout per CDNA5 ISA 7.12.2:
//   C/D: vgpr j = row (j + 8*half), col lm          (half = lane>>4, lm = lane&15)
//   A 16x4: lanes 0-15 K={k0,k0+1}, lanes 16-31 K={k0+2,k0+3}, row = lm
//   B 4x16: mirrored (row K striped across lanes within a VGPR)
// ---------------------------------------------------------------------------
__device__ __forceinline__ v8f gemm_tile(const float* __restrict__ A, int lda,
                                         const float* __restrict__ W, int ldw,
                                         int n0, int K, float cinit, int lane)
{
    const int half = lane >> 4;
    const int lm   = lane & 15;
    v8f c;
#pragma unroll
    for (int j = 0; j < 8; ++j) c[j] = cinit;
    for (int k0 = 0; k0 < K; k0 += 4) {
        const int ka = k0 + 2 * half;
        v2f a, b;
        a[0] = A[lm * lda + ka];
        a[1] = A[lm * lda + ka + 1];
        b[0] = W[ka * ldw + n0 + lm];
        b[1] = W[(ka + 1) * ldw + n0 + lm];
        c = __builtin_amdgcn_wmma_f32_16x16x4_f32(false, a, false, b,
                                                  (short)0, c, false, false);
    }
    return c;
}

// stage a 16x64 fp32 tile from global (row-major, ld=64) into LDS with stride 68
__device__ __forceinline__ void stage_tile(const float* __restrict__ g, int r0,
                                           float* __restrict__ lds, int lane)
{
#pragma unroll
    for (int i = lane; i < 16 * 16; i += 32) {
        const int r = i >> 4, q = i & 15;
        const float4 v = *(const float4*)(g + (size_t)(r0 + r) * D + q * 4);
        *(float4*)(lds + r * 68 + q * 4) = v;
    }
}

// ---------------------------------------------------------------------------
// Encoder: x = relu(relu(nodes @ w0 + b0) @ w1 + b1)
// ---------------------------------------------------------------------------
__global__ __launch_bounds__(128)
void gcn_encoder(const float* __restrict__ nodes,
                 const float* __restrict__ w0, const float* __restrict__ b0,
                 const float* __restrict__ w1, const float* __restrict__ b1,
                 float* __restrict__ xout)
{
    __shared__ float sW0[D * D];
    __shared__ float sW1[D * D];
    __shared__ float sA[4][16 * 68];
    __shared__ float sH[4][16 * 68];

    const int tid = threadIdx.x;
    for (int i = tid; i < D * D; i += 128) { sW0[i] = w0[i]; sW1[i] = w1[i]; }

    const int wave = tid >> 5;
    const int lane = tid & 31;
    const int tile = blockIdx.x * 4 + wave;
    const int half = lane >> 4;
    const int lm   = lane & 15;
    float* A = sA[wave];
    float* H = sH[wave];

    if (tile < TILES) stage_tile(nodes, tile * 16, A, lane);
    __syncthreads();

    if (tile < TILES) {
        for (int n0 = 0; n0 < D; n0 += 16) {
            v8f c = gemm_tile(A, 68, sW0, D, n0, D, b0[n0 + lm], lane);
#pragma unroll
            for (int j = 0; j < 8; ++j)
                H[(j + 8 * half) * 68 + n0 + lm] = fmaxf(c[j], 0.f);
        }
    }
    __syncthreads();

    if (tile < TILES) {
        const int r0 = tile * 16;
        for (int n0 = 0; n0 < D; n0 += 16) {
            v8f c = gemm_tile(H, 68, sW1, D, n0, D, b1[n0 + lm], lane);
#pragma unroll
            for (int j = 0; j < 8; ++j)
                xout[(size_t)(r0 + j + 8 * half) * D + n0 + lm] = fmaxf(c[j], 0.f);
        }
    }
}

// ---------------------------------------------------------------------------
// conv = 0
// ---------------------------------------------------------------------------
__global__ void gcn_zero(float4* __restrict__ p, int n4)
{
    const int i = blockIdx.x * blockDim.x + threadIdx.x;
    if (i < n4) p[i] = make_float4(0.f, 0.f, 0.f, 0.f);
}

// ---------------------------------------------------------------------------
// conv[senders[e]] += edges[e] * x[receivers[e]]   (16 lanes per edge, f32
// atomics resolved at L2; x and conv both fit in the 192 MB L2)
// ---------------------------------------------------------------------------
__global__ __launch_bounds__(256)
void gcn_edge_scatter(const float* __restrict__ x,
                      const float* __restrict__ ew,
                      const int*   __restrict__ senders,
                      const int*   __restrict__ receivers,
                      float* __restrict__ conv)
{
    const long long tid = (long long)blockIdx.x * blockDim.x + threadIdx.x;
    const int e = (int)(tid >> 4);
    const int q = (int)(tid & 15);
    if (e >= N_EDGES) return;

    const float w = ew[e];
    const int   r = receivers[e];
    const int   s = senders[e];
    const float4 v = *(const float4*)(x + (size_t)r * D + q * 4);
    float* dst = conv + (size_t)s * D + q * 4;
    unsafeAtomicAdd(dst + 0, w * v.x);
    unsafeAtomicAdd(dst + 1, w * v.y);
    unsafeAtomicAdd(dst + 2, w * v.z);
    unsafeAtomicAdd(dst + 3, w * v.w);
}

// ---------------------------------------------------------------------------
// x = relu(conv @ w + b) + conv
// ---------------------------------------------------------------------------
__global__ __launch_bounds__(128)
void gcn_update(const float* __restrict__ conv,
                const float* __restrict__ w, const float* __restrict__ b,
                float* __restrict__ xout)
{
    __shared__ float sW[D * D];
    __shared__ float sA[4][16 * 68];

    const int tid = threadIdx.x;
    for (int i = tid; i < D * D; i += 128) sW[i] = w[i];

    const int wave = tid >> 5;
    const int lane = tid & 31;
    const int tile = blockIdx.x * 4 + wave;
    const int half = lane >> 4;
    const int lm   = lane & 15;
    float* A = sA[wave];

    if (tile < TILES) stage_tile(conv, tile * 16, A, lane);
    __syncthreads();

    if (tile < TILES) {
        const int r0 = tile * 16;
        for (int n0 = 0; n0 < D; n0 += 16) {
            v8f c = gemm_tile(A, 68, sW, D, n0, D, b[n0 + lm], lane);
#pragma unroll
            for (int j = 0; j < 8; ++j) {
                const float skip = A[(j + 8 * half) * 68 + n0 + lm];
                xout[(size_t)(r0 + j + 8 * half) * D + n0 + lm] =
                    fmaxf(c[j], 0.f) + skip;
            }
        }
    }
}

// ---------------------------------------------------------------------------
// Decoder: out = relu(x @ w0 + b0) @ w1 + b1   ([N,40])
// w1 is zero-padded to stride 48 in LDS so the n0=32 tile reads in-bounds.
// ---------------------------------------------------------------------------
__global__ __launch_bounds__(128)
void gcn_decoder(const float* __restrict__ x,
                 const float* __restrict__ w0, const float* __restrict__ b0,
                 const float* __restrict__ w1, const float* __restrict__ b1,
                 float* __restrict__ out)
{
    __shared__ float sW0[D * D];
    __shared__ float sW1[D * 48];
    __shared__ float sA[4][16 * 68];
    __shared__ float sH[4][16 * 68];

    const int tid = threadIdx.x;
    for (int i = tid; i < D * D; i += 128) sW0[i] = w0[i];
    for (int i = tid; i < D * 48; i += 128) {
        const int rr = i / 48, cc = i % 48;
        sW1[i] = (cc < N_CLASSES) ? w1[rr * N_CLASSES + cc] : 0.f;
    }

    const int wave = tid >> 5;
    const int lane = tid & 31;
    const int tile = blockIdx.x * 4 + wave;
    const int half = lane >> 4;
    const int lm   = lane & 15;
    float* A = sA[wave];
    float* H = sH[wave];

    if (tile < TILES) stage_tile(x, tile * 16, A, lane);
    __syncthreads();

    if (tile < TILES) {
        for (int n0 = 0; n0 < D; n0 += 16) {
            v8f c = gemm_tile(A, 68, sW0, D, n0, D, b0[n0 + lm], lane);
#pragma unroll
            for (int j = 0; j < 8; ++j)
                H[(j + 8 * half) * 68 + n0 + lm] = fmaxf(c[j], 0.f);
        }
    }
    __syncthreads();

    if (tile < TILES) {
        const int r0 = tile * 16;
        for (int n0 = 0; n0 < N_CLASSES; n0 += 16) {
            const bool valid = (n0 + lm) < N_CLASSES;
            const float bias = valid ? b1[n0 + lm] : 0.f;
            v8f c = gemm_tile(H, 68, sW1, 48, n0, D, bias, lane);
            if (valid) {
#pragma unroll
                for (int j = 0; j < 8; ++j)
                    out[(size_t)(r0 + j + 8 * half) * N_CLASSES + n0 + lm] = c[j];
            }
        }
    }
}

// ---------------------------------------------------------------------------
extern "C" void kernel_launch(void* const* d_in, const int* in_sizes, int n_in,
                              void* d_out, int out_size, void* d_ws, size_t ws_size,
                              hipStream_t stream)
{
    const float* nodes     = (const float*)d_in[0];
    const float* edges     = (const float*)d_in[1];
    const int*   senders   = (const int*)d_in[2];
    const int*   receivers = (const int*)d_in[3];
    const float* enc_w0 = (const float*)d_in[4];
    const float* enc_b0 = (const float*)d_in[5];
    const float* enc_w1 = (const float*)d_in[6];
    const float* enc_b1 = (const float*)d_in[7];
    const float* core_w[2] = { (const float*)d_in[8],  (const float*)d_in[10] };
    const float* core_b[2] = { (const float*)d_in[9],  (const float*)d_in[11] };
    const float* dec_w0 = (const float*)d_in[12];
    const float* dec_b0 = (const float*)d_in[13];
    const float* dec_w1 = (const float*)d_in[14];
    const float* dec_b1 = (const float*)d_in[15];
    float* out = (float*)d_out;

    float* xbuf    = (float*)d_ws;                       // N*64 f32
    float* convbuf = xbuf + (size_t)N_NODES * D;         // N*64 f32

    const dim3 gT((TILES + 3) / 4), bT(128);
    const int  n4 = N_NODES * D / 4;

    gcn_encoder<<<gT, bT, 0, stream>>>(nodes, enc_w0, enc_b0, enc_w1, enc_b1, xbuf);

    for (int r = 0; r < 2; ++r) {
        gcn_zero<<<(n4 + 255) / 256, 256, 0, stream>>>((float4*)convbuf, n4);
        gcn_edge_scatter<<<(N_EDGES * 16 + 255) / 256, 256, 0, stream>>>(
            xbuf, edges, senders, receivers, convbuf);
        gcn_update<<<gT, bT, 0, stream>>>(convbuf, core_w[r], core_b[r], xbuf);
    }

    gcn_decoder<<<gT, bT, 0, stream>>>(xbuf, dec_w0, dec_b0, dec_w1, dec_b1, out);
}